// CTMBlock_34651796144713
// MI455X (gfx1250) — compile-verified
//
#include <hip/hip_runtime.h>
#include <math.h>

// ---------------------------------------------------------------------------
// CTMBlock for MI455X (gfx1250). Channel-mixing ops are fp32 WMMA
// (V_WMMA_F32_16X16X4_F32). 1x1-conv GEMMs use the CDNA5 pipeline:
// TDM tensor_load_to_lds stages the KxN activation panel into LDS once per
// workgroup, then all M-tiles are computed from LDS (8-16x less global X
// traffic; workload is memory-bound: AI ~32 FLOP/B << 745 FLOP/B needed to
// saturate WMMA at 23.3 TB/s).
// ---------------------------------------------------------------------------

typedef __attribute__((ext_vector_type(2))) float    v2f;
typedef __attribute__((ext_vector_type(8))) float    v8f;
typedef __attribute__((ext_vector_type(4))) unsigned v4u;
typedef __attribute__((ext_vector_type(8))) int      v8i;
typedef __attribute__((ext_vector_type(4))) int      v4i;

constexpr int  HWN  = 16384;         // 128*128 pixels
constexpr long S64  = 8L * 64 * HWN; // (8,64,HW) floats
constexpr long S128 = 8L * 128 * HWN;
constexpr long S256 = 8L * 256 * HWN;

__device__ __forceinline__ v8f wmma_k4(v2f a, v2f b, v8f c) {
  // D = A(16x4) * B(4x16) + C(16x16), fp32
  return __builtin_amdgcn_wmma_f32_16x16x4_f32(
      /*neg_a=*/false, a, /*neg_b=*/false, b,
      /*c_mod=*/(short)0, c, /*reuse_a=*/false, /*reuse_b=*/false);
}

#define HAVE_TDM (__has_builtin(__builtin_amdgcn_tensor_load_to_lds) && \
                  __has_builtin(__builtin_amdgcn_s_wait_tensorcnt) &&   \
                  __has_builtin(__builtin_amdgcn_groupstaticsize))

// ---------------------------------------------------------------------------
// LDS-staged per-batch GEMM: Y[b] (MxP) = W (MxK, row-major) * X[b] (KxP)
// One WG per 64-pixel column block: TDM DMAs the K x 64 X-panel into LDS
// (rows contiguous), then loops all M/16 tiles against it.
// grid: (P/64, 1, B), block 128 (4 waves; wave w -> 16-pixel sub-tile w)
// dynamic LDS: K*64*4 bytes
// ---------------------------------------------------------------------------
__global__ __launch_bounds__(128)
void gemm_lds_k(const float* __restrict__ W, const float* __restrict__ X,
                float* __restrict__ Y, const float* __restrict__ bias,
                const float* __restrict__ resid,
                int M, int K, long sX, long sY, long sR)
{
  extern __shared__ float tile[];   // [K][64]
  const int P = HWN;
  int wave = threadIdx.x >> 5;
  int lane = threadIdx.x & 31;
  int col  = lane & 15;
  int kh   = lane >> 4;
  int n0   = blockIdx.x * 64;
  int b    = blockIdx.z;
  const float* Xb = X + (long)b * sX + n0;

#if HAVE_TDM
  if (threadIdx.x < 32) {  // wave 0 issues the tensor DMA
    unsigned long long ga = (unsigned long long)(const void*)Xb;
    unsigned lds_base = (unsigned)__builtin_amdgcn_groupstaticsize();
    v4u g0; v8i g1; v4i g2 = {0, 0, 0, 0}, g3 = {0, 0, 0, 0};
    // D# group0: count=1 (valid), lds_addr, global_addr[56:0], type=2
    g0[0] = 1u;
    g0[1] = lds_base;
    g0[2] = (unsigned)(ga & 0xffffffffu);
    g0[3] = (unsigned)((ga >> 32) & 0x1ffffffu) | (2u << 30);
    // D# group1: data_size=2 (4B); tensor_dim0=HW, tensor_dim1=K;
    // tile_dim0=64 (pixels), tile_dim1=K rows; dim0 stride = HW elements.
    g1[0] = (int)(2u << 16);
    g1[1] = (int)(((unsigned)HWN & 0xffffu) << 16);
    g1[2] = (int)(((unsigned)HWN >> 16) | (((unsigned)K & 0xffffu) << 16));
    g1[3] = (int)(((unsigned)K >> 16) | (64u << 16));
    g1[4] = (int)((unsigned)K & 0xffffu);      // tile_dim1=K, tile_dim2=0
    g1[5] = (int)(unsigned)HWN;                // tensor_dim0_stride lo32
    g1[6] = 0;
    g1[7] = 0;
#if defined(__clang_major__) && (__clang_major__ >= 23)
    v8i g4 = {0, 0, 0, 0, 0, 0, 0, 0};
    __builtin_amdgcn_tensor_load_to_lds(g0, g1, g2, g3, g4, 0);
#else
    __builtin_amdgcn_tensor_load_to_lds(g0, g1, g2, g3, 0);
#endif
    __builtin_amdgcn_s_wait_tensorcnt(0);
  }
  __syncthreads();
#else
  for (int i = threadIdx.x; i < K * 64; i += 128)
    tile[i] = Xb[(long)(i >> 6) * P + (i & 63)];
  __syncthreads();
#endif

  float* Yb = Y + (long)b * sY;
  int nw = wave * 16 + col;           // pixel within 64-block
  for (int m0 = 0; m0 < M; m0 += 16) {
    const float* Wr = W + (long)(m0 + col) * K;
    if (m0 + 16 < M)
      __builtin_prefetch(W + (long)(m0 + 16 + col) * K, 0, 0);
    v8f acc = {};
    for (int k0 = 0; k0 < K; k0 += 4) {
      int ka = k0 + 2 * kh;
      v2f a, bf;
      a.x  = Wr[ka];
      a.y  = Wr[ka + 1];
      bf.x = tile[ka * 64 + nw];
      bf.y = tile[(ka + 1) * 64 + nw];
      acc = wmma_k4(a, bf, acc);
    }
    #pragma unroll
    for (int j = 0; j < 8; ++j) {
      int m = m0 + j + 8 * kh;
      long idx = (long)m * P + n0 + nw;
      float v = acc[j];
      if (bias)  v += bias[m];
      if (resid) v += resid[(long)b * sR + idx];
      Yb[idx] = v;
    }
  }
}

// ---------------------------------------------------------------------------
// Dense 3x3 conv, 64->64 ch, pad 1, as 9 shifted K=64 GEMMs (WMMA).
// act: 0 none, 1 leaky_relu(0.2). Optional residual.
// grid: (HW/64, 4, B), block 128
// ---------------------------------------------------------------------------
__global__ __launch_bounds__(128)
void conv3_k(const float* __restrict__ Wt, const float* __restrict__ X,
             float* __restrict__ Y, const float* __restrict__ bias,
             const float* __restrict__ resid, long sX, long sY, long sR, int act)
{
  const int P = HWN;
  int wave = threadIdx.x >> 5;
  int lane = threadIdx.x & 31;
  int col = lane & 15, kh = lane >> 4;
  int n0 = blockIdx.x * 64 + wave * 16;
  int m0 = blockIdx.y * 16;
  int b  = blockIdx.z;
  const float* Xb = X + (long)b * sX;
  int p  = n0 + col;
  int ph = p >> 7, pw = p & 127;
  v8f acc = {};
  for (int tap = 0; tap < 9; ++tap) {
    int dy = tap / 3 - 1, dx = tap % 3 - 1;
    int hh = ph + dy, ww = pw + dx;
    bool ok = ((unsigned)hh < 128u) && ((unsigned)ww < 128u);
    long poff = ok ? (long)(hh * 128 + ww) : 0;
    const float* Wtap = Wt + (long)(m0 + col) * 64 * 9 + tap;
    for (int c0 = 0; c0 < 64; c0 += 4) {
      int ka = c0 + 2 * kh;
      v2f a, bf;
      a.x = Wtap[(long)ka * 9];
      a.y = Wtap[(long)(ka + 1) * 9];
      bf.x = ok ? Xb[(long)ka * P + poff] : 0.f;
      bf.y = ok ? Xb[(long)(ka + 1) * P + poff] : 0.f;
      acc = wmma_k4(a, bf, acc);
    }
  }
  float* Yb = Y + (long)b * sY;
  #pragma unroll
  for (int j = 0; j < 8; ++j) {
    int m = m0 + j + 8 * kh;
    long idx = (long)m * P + n0 + col;
    float v = acc[j] + bias[m];
    if (act == 1) v = v > 0.f ? v : 0.2f * v;
    if (resid) v += resid[(long)b * sR + idx];
    Yb[idx] = v;
  }
}

// ---------------------------------------------------------------------------
// attn partial: per (chunk, head, b): 16x16 += Kh[:,n] * Qh[:,n]^T over 512 n.
// Reduction dim n -> WMMA K. grid (32,4,8), block 32 (one wave).
// ---------------------------------------------------------------------------
__global__ __launch_bounds__(32)
void attn_part_k(const float* __restrict__ q, const float* __restrict__ kmat,
                 float* __restrict__ part)
{
  int lane = threadIdx.x & 31;
  int col = lane & 15, kh = lane >> 4;
  int chunk = blockIdx.x, head = blockIdx.y, b = blockIdx.z;
  const float* Kb = kmat + ((long)b * 64 + head * 16 + col) * HWN;
  const float* Qb = q    + ((long)b * 64 + head * 16 + col) * HWN;
  int nbase = chunk * 512;
  v8f acc = {};
  for (int t = 0; t < 128; ++t) {
    int nb = nbase + t * 4 + 2 * kh;
    v2f a, bf;
    a.x = Kb[nb];  a.y = Kb[nb + 1];   // A: rows d (lane col), K = n
    bf.x = Qb[nb]; bf.y = Qb[nb + 1];  // B: cols e (lane col), K = n
    acc = wmma_k4(a, bf, acc);
  }
  long base = (((long)(b * 4 + head)) * 32 + chunk) * 256;
  #pragma unroll
  for (int j = 0; j < 8; ++j)
    part[base + (j + 8 * kh) * 16 + col] = acc[j];
}

// Deterministic fixed-order reduction of 32 chunk-partials, * rescale[head]
__global__ void attn_reduce_k(const float* __restrict__ part,
                              const float* __restrict__ rescale,
                              float* __restrict__ attn)
{
  int bh = blockIdx.x, t = threadIdx.x;  // t in [0,256)
  float s = 0.f;
  for (int c = 0; c < 32; ++c) s += part[((long)bh * 32 + c) * 256 + t];
  attn[(long)bh * 256 + t] = s * rescale[bh & 3];
}

__global__ void softmax16_k(float* __restrict__ attn)
{
  int bh = blockIdx.x, d = threadIdx.x;
  if (d < 16) {
    float* row = attn + (long)bh * 256 + d * 16;
    float mx = row[0];
    for (int e = 1; e < 16; ++e) mx = fmaxf(mx, row[e]);
    float ex[16], s = 0.f;
    for (int e = 0; e < 16; ++e) { ex[e] = expf(row[e] - mx); s += ex[e]; }
    float inv = 1.f / s;
    for (int e = 0; e < 16; ++e) row[e] = ex[e] * inv;
  }
}

// xo[d,n] = sum_e attn[d,e] * V[e,n]   (M=16,K=16, WMMA) grid (HW/64,4,8)
__global__ __launch_bounds__(128)
void xo_k(const float* __restrict__ attn, const float* __restrict__ v,
          float* __restrict__ xo)
{
  int wave = threadIdx.x >> 5, lane = threadIdx.x & 31;
  int col = lane & 15, kh = lane >> 4;
  int n0 = blockIdx.x * 64 + wave * 16;
  int head = blockIdx.y, b = blockIdx.z;
  const float* A  = attn + (long)(b * 4 + head) * 256;
  const float* Vb = v + ((long)b * 64 + head * 16) * HWN;
  v8f acc = {};
  #pragma unroll
  for (int k0 = 0; k0 < 16; k0 += 4) {
    int ka = k0 + 2 * kh;
    v2f a, bf;
    a.x = A[col * 16 + ka];
    a.y = A[col * 16 + ka + 1];
    bf.x = Vb[(long)ka * HWN + n0 + col];
    bf.y = Vb[(long)(ka + 1) * HWN + n0 + col];
    acc = wmma_k4(a, bf, acc);
  }
  float* Ob = xo + ((long)b * 64 + head * 16) * HWN;
  #pragma unroll
  for (int j = 0; j < 8; ++j)
    Ob[(long)(j + 8 * kh) * HWN + n0 + col] = acc[j];
}

// ---------------------------------------------------------------------------
// row L2-normalize in place: 512 rows of HW (q or k). One WG per row.
// ---------------------------------------------------------------------------
__global__ __launch_bounds__(256)
void l2norm_k(float* __restrict__ x)
{
  __shared__ float red[256];
  long base = (long)blockIdx.x * HWN;
  float s = 0.f;
  for (int i = threadIdx.x; i < HWN; i += 256) { float v = x[base + i]; s += v * v; }
  red[threadIdx.x] = s;
  __syncthreads();
  for (int st = 128; st > 0; st >>= 1) {
    if (threadIdx.x < st) red[threadIdx.x] += red[threadIdx.x + st];
    __syncthreads();
  }
  float inv = 1.f / fmaxf(sqrtf(red[0]), 1e-12f);
  for (int i = threadIdx.x; i < HWN; i += 256) x[base + i] *= inv;
}

// per-(b,c) mean & max over HW (channel pools). 512 WGs.
__global__ __launch_bounds__(256)
void chpool_k(const float* __restrict__ x, float* __restrict__ ap,
              float* __restrict__ mp)
{
  __shared__ float rs[256], rm[256];
  long base = (long)blockIdx.x * HWN;
  float s = 0.f, m = -3.402823466e38f;
  for (int i = threadIdx.x; i < HWN; i += 256) {
    float v = x[base + i]; s += v; m = fmaxf(m, v);
  }
  rs[threadIdx.x] = s; rm[threadIdx.x] = m;
  __syncthreads();
  for (int st = 128; st > 0; st >>= 1) {
    if (threadIdx.x < st) {
      rs[threadIdx.x] += rs[threadIdx.x + st];
      rm[threadIdx.x] = fmaxf(rm[threadIdx.x], rm[threadIdx.x + st]);
    }
    __syncthreads();
  }
  if (threadIdx.x == 0) {
    ap[blockIdx.x] = rs[0] * (1.f / HWN);
    mp[blockIdx.x] = rm[0];
  }
}

// per-pixel mean & max over 64 channels. grid (HW/256, 8)
__global__ void spa_stage_k(const float* __restrict__ o, float* __restrict__ savg,
                            float* __restrict__ smax)
{
  int b = blockIdx.y;
  int p = blockIdx.x * 256 + threadIdx.x;
  const float* xb = o + (long)b * 64 * HWN + p;
  float s = 0.f, m = -3.402823466e38f;
  for (int c = 0; c < 64; ++c) { float v = xb[(long)c * HWN]; s += v; m = fmaxf(m, v); }
  savg[(long)b * HWN + p] = s * (1.f / 64.f);
  smax[(long)b * HWN + p] = m;
}

// 7x7 conv over {avg,max} 2ch -> sigmoid. grid (HW/256, 8)
__global__ void spa_conv_k(const float* __restrict__ savg,
                           const float* __restrict__ smax,
                           const float* __restrict__ w, float* __restrict__ spa)
{
  int b = blockIdx.y;
  int p = blockIdx.x * 256 + threadIdx.x;
  int h = p >> 7, ww = p & 127;
  float s = 0.f;
  for (int ky = 0; ky < 7; ++ky) {
    int hh = h + ky - 3;
    if ((unsigned)hh >= 128u) continue;
    for (int kx = 0; kx < 7; ++kx) {
      int wx = ww + kx - 3;
      if ((unsigned)wx >= 128u) continue;
      long q = (long)b * HWN + hh * 128 + wx;
      s += w[ky * 7 + kx] * savg[q] + w[49 + ky * 7 + kx] * smax[q];
    }
  }
  spa[(long)b * HWN + p] = 1.f / (1.f + expf(-s));
}

// tiny channel-MLP + grouped (2->1) conv -> sigmoid spe[b][c]. grid 8, block 64
__global__ __launch_bounds__(64)
void spe_k(const float* __restrict__ ap, const float* __restrict__ mp,
           const float* __restrict__ m1w, const float* __restrict__ m1b,
           const float* __restrict__ m2w, const float* __restrict__ m2b,
           const float* __restrict__ c21w, const float* __restrict__ c21b,
           float* __restrict__ spe)
{
  __shared__ float hid[8];
  int b = blockIdx.x, t = threadIdx.x;
  if (t < 8) {
    int j = t & 3;
    const float* pool = (t < 4) ? (mp + b * 64) : (ap + b * 64);
    float s = m1b[j];
    for (int c = 0; c < 64; ++c) s += m1w[j * 64 + c] * pool[c];
    hid[t] = fmaxf(s, 0.f);
  }
  __syncthreads();
  int c = t;
  float om = m2b[c], oa = m2b[c];
  for (int j = 0; j < 4; ++j) {
    om += m2w[c * 4 + j] * hid[j];
    oa += m2w[c * 4 + j] * hid[4 + j];
  }
  float s = c21w[c * 2 + 0] * om + c21w[c * 2 + 1] * oa + c21b[c];
  spe[b * 64 + c] = 1.f / (1.f + expf(-s));
}

// conv_x_new = out * sigmoid(spa + spe) + conv_x (in place on y[:, :64])
__global__ void applyatt_k(float* __restrict__ y, const float* __restrict__ o,
                           const float* __restrict__ spa,
                           const float* __restrict__ spe)
{
  long i = (long)blockIdx.x * 256 + threadIdx.x;  // over 8*64*HW
  long p = i % HWN; long r = i / HWN;
  int c = (int)(r & 63); int b = (int)(r >> 6);
  float att = 1.f / (1.f + expf(-(spa[(long)b * HWN + p] + spe[b * 64 + c])));
  long yi = ((long)b * 128 + c) * HWN + p;
  y[yi] = o[((long)b * 64 + c) * HWN + p] * att + y[yi];
}

// LayerNorm over 64 channels per pixel. grid (HW/256, 8)
__global__ void ln_k(const float* __restrict__ t, const float* __restrict__ g,
                     const float* __restrict__ be, float* __restrict__ o)
{
  int b = blockIdx.y;
  int p = blockIdx.x * 256 + threadIdx.x;
  const float* xb = t + (long)b * 64 * HWN + p;
  float s = 0.f, s2 = 0.f;
  for (int c = 0; c < 64; ++c) { float v = xb[(long)c * HWN]; s += v; s2 += v * v; }
  float m = s * (1.f / 64.f);
  float var = s2 * (1.f / 64.f) - m * m;
  float inv = rsqrtf(var + 1e-5f);
  float* ob = o + (long)b * 64 * HWN + p;
  for (int c = 0; c < 64; ++c)
    ob[(long)c * HWN] = (xb[(long)c * HWN] - m) * inv * g[c] + be[c];
}

// depthwise conv (ks in {1,3,5,7}), 64 channels, optional gelu / 2 adds
// grid (HW/256, 64, 8)
__global__ void dw_k(const float* __restrict__ X, const float* __restrict__ Wt,
                     float* __restrict__ Y, const float* __restrict__ add0,
                     const float* __restrict__ add1,
                     int ks, int pad, int gelu,
                     long sX, long sY, long sA0, long sA1)
{
  int c = blockIdx.y, b = blockIdx.z;
  int p = blockIdx.x * 256 + threadIdx.x;
  int h = p >> 7, w = p & 127;
  const float* xp = X + (long)b * sX + (long)c * HWN;
  float s = 0.f;
  for (int ky = 0; ky < ks; ++ky) {
    int hh = h + ky - pad;
    if ((unsigned)hh >= 128u) continue;
    for (int kx = 0; kx < ks; ++kx) {
      int ww = w + kx - pad;
      if ((unsigned)ww >= 128u) continue;
      s += Wt[(c * ks + ky) * ks + kx] * xp[hh * 128 + ww];
    }
  }
  if (gelu) s = 0.5f * s * (1.f + erff(s * 0.70710678118654752f));
  long pi = (long)c * HWN + p;
  if (add0) s += add0[(long)b * sA0 + pi];
  if (add1) s += add1[(long)b * sA1 + pi];
  Y[(long)b * sY + pi] = s;
}

// ---------------------------------------------------------------------------
extern "C" void kernel_launch(void* const* d_in, const int* in_sizes, int n_in,
                              void* d_out, int out_size, void* d_ws, size_t ws_size,
                              hipStream_t stream)
{
  (void)in_sizes; (void)n_in; (void)out_size; (void)ws_size;
  const float* x     = (const float*)d_in[0];
  const float* in_w  = (const float*)d_in[1];
  const float* in_b  = (const float*)d_in[2];
  const float* out_w = (const float*)d_in[3];
  const float* out_b = (const float*)d_in[4];
  const float* c1w   = (const float*)d_in[5];
  const float* c1b   = (const float*)d_in[6];
  const float* c2w   = (const float*)d_in[7];
  const float* c2b   = (const float*)d_in[8];
  const float* spa_w = (const float*)d_in[9];
  const float* m1w   = (const float*)d_in[10];
  const float* m1b   = (const float*)d_in[11];
  const float* m2w   = (const float*)d_in[12];
  const float* m2b   = (const float*)d_in[13];
  const float* c21w  = (const float*)d_in[14];
  const float* c21b  = (const float*)d_in[15];
  const float* qw    = (const float*)d_in[16];
  const float* kw    = (const float*)d_in[17];
  const float* vw    = (const float*)d_in[18];
  const float* rescale = (const float*)d_in[19];
  const float* pw    = (const float*)d_in[20];
  const float* pb    = (const float*)d_in[21];
  const float* pos1  = (const float*)d_in[22];
  const float* pos2  = (const float*)d_in[23];
  const float* ln_g  = (const float*)d_in[24];
  const float* ln_b  = (const float*)d_in[25];
  const float* fw1   = (const float*)d_in[26];
  const float* fw2   = (const float*)d_in[27];
  const float* dw1   = (const float*)d_in[28];
  const float* dw3   = (const float*)d_in[29];
  const float* dw5   = (const float*)d_in[30];
  const float* dw7   = (const float*)d_in[31];

  float* ws = (float*)d_ws;
  // workspace layout (floats); FFN hidden scratch mm1 aliases [q..t0) = S256
  float* y   = ws;              // (8,128,HW)  in-proj out -> final concat
  float* t1  = y  + S128;       // (8,64,HW)
  float* qb  = t1 + S64;        // (8,64,HW)   q, later xo
  float* kb  = qb + S64;
  float* vb  = kb + S64;
  float* t0  = vb + S64;        // (8,64,HW)
  float* m0  = t0 + S64;        // (8,256,HW)  FFN hidden
  float* mm1 = qb;              // alias: (8,256,HW) over q..t0 (exactly S256)
  float* st  = m0 + S256;       // small stats
  float* savg = st;             // 8*HW
  float* smax = savg + 8L * HWN;
  float* spa  = smax + 8L * HWN;
  float* ap   = spa + 8L * HWN; // 512
  float* mp   = ap + 512;
  float* spe  = mp + 512;
  float* part = spe + 512;      // 8*4*32*256
  float* attn = part + 262144;  // 8*4*256

  const long s64 = 64L * HWN, s128 = 128L * HWN, s256 = 256L * HWN;
  const long co64 = 64L * HWN;  // channel offset within y
  dim3 blk(128);
  dim3 ggrid(256, 1, 8);
  const int shm64  = 64  * 64 * 4;   // K=64  panel
  const int shm128 = 128 * 64 * 4;   // K=128 panel
  const int shm256 = 256 * 64 * 4;   // K=256 panel

  // ---- in-projection: y = in_w * x + in_b  (M=128,K=128) ----
  gemm_lds_k<<<ggrid, blk, shm128, stream>>>(in_w, x, y, in_b, nullptr, 128, 128, s128, s128, 0);

  // ---- CNN branch (conv_x = y[:, :64]) ----
  conv3_k<<<dim3(256, 4, 8), blk, 0, stream>>>(c1w, y, t0, c1b, nullptr, s128, s64, 0, 1);
  conv3_k<<<dim3(256, 4, 8), blk, 0, stream>>>(c2w, t0, t1, c2b, y, s64, s64, s128, 0);
  spa_stage_k<<<dim3(64, 8), 256, 0, stream>>>(t1, savg, smax);
  spa_conv_k<<<dim3(64, 8), 256, 0, stream>>>(savg, smax, spa_w, spa);
  chpool_k<<<512, 256, 0, stream>>>(t1, ap, mp);
  spe_k<<<8, 64, 0, stream>>>(ap, mp, m1w, m1b, m2w, m2b, c21w, c21b, spe);
  applyatt_k<<<(int)(S64 / 256), 256, 0, stream>>>(y, t1, spa, spe);

  // ---- transformer branch (trans_x = y[:, 64:]) ----
  gemm_lds_k<<<ggrid, blk, shm64, stream>>>(qw, y + co64, qb, nullptr, nullptr, 64, 64, s128, s64, 0);
  gemm_lds_k<<<ggrid, blk, shm64, stream>>>(kw, y + co64, kb, nullptr, nullptr, 64, 64, s128, s64, 0);
  gemm_lds_k<<<ggrid, blk, shm64, stream>>>(vw, y + co64, vb, nullptr, nullptr, 64, 64, s128, s64, 0);
  l2norm_k<<<512, 256, 0, stream>>>(qb);
  l2norm_k<<<512, 256, 0, stream>>>(kb);
  attn_part_k<<<dim3(32, 4, 8), 32, 0, stream>>>(qb, kb, part);
  attn_reduce_k<<<32, 256, 0, stream>>>(part, rescale, attn);
  softmax16_k<<<32, 32, 0, stream>>>(attn);
  xo_k<<<dim3(256, 4, 8), blk, 0, stream>>>(attn, vb, qb);  // xo -> qb
  gemm_lds_k<<<ggrid, blk, shm64, stream>>>(pw, qb, t1, pb, nullptr, 64, 64, s64, s64, 0);
  // pos path: t0 = gelu(dw3(v)); t1 += dw3(t0) + trans_x (full MSA residual)
  dw_k<<<dim3(64, 64, 8), 256, 0, stream>>>(vb, pos1, t0, nullptr, nullptr, 3, 1, 1, s64, s64, 0, 0);
  dw_k<<<dim3(64, 64, 8), 256, 0, stream>>>(t0, pos2, t1, t1, y + co64, 3, 1, 0, s64, s64, s64, s128);
  // LN -> t0 ; FFN
  ln_k<<<dim3(64, 8), 256, 0, stream>>>(t1, ln_g, ln_b, t0);
  gemm_lds_k<<<ggrid, blk, shm64, stream>>>(fw1, t0, m0, nullptr, nullptr, 256, 64, s64, s256, 0);
  dw_k<<<dim3(64, 64, 8), 256, 0, stream>>>(m0,            dw1, mm1,            m0,            nullptr, 1, 0, 0, s256, s256, s256, 0);
  dw_k<<<dim3(64, 64, 8), 256, 0, stream>>>(m0 + co64,     dw3, mm1 + co64,     m0 + co64,     nullptr, 3, 1, 0, s256, s256, s256, 0);
  dw_k<<<dim3(64, 64, 8), 256, 0, stream>>>(m0 + 2 * co64, dw5, mm1 + 2 * co64, m0 + 2 * co64, nullptr, 5, 2, 0, s256, s256, s256, 0);
  dw_k<<<dim3(64, 64, 8), 256, 0, stream>>>(m0 + 3 * co64, dw7, mm1 + 3 * co64, m0 + 3 * co64, nullptr, 7, 3, 0, s256, s256, s256, 0);
  // trans_x_new = fw2 * o + t1  -> y[:, 64:]
  gemm_lds_k<<<ggrid, blk, shm256, stream>>>(fw2, mm1, y + co64, nullptr, t1, 64, 256, s256, s128, s64);

  // ---- out-projection + outer residual: out = out_w * y + out_b + x ----
  gemm_lds_k<<<ggrid, blk, shm128, stream>>>(out_w, y, (float*)d_out, out_b, x, 128, 128, s128, s128, s128);
}